// RadarNet_61254823575567
// MI455X (gfx1250) — compile-verified
//
#include <hip/hip_runtime.h>
#include <hip/hip_bf16.h>

#define WIN_      20
#define TLEN_     512
#define NWIN_     493          // (512-20)/1 + 1
#define NPAD_     496          // zero-padded to multiple of 4 for WMMA K-steps
#define D1_       16
#define D2_       8
#define NCLS_     3
#define REEIG_EPS_ 1e-4f

typedef __attribute__((ext_vector_type(2))) float v2f;
typedef __attribute__((ext_vector_type(8))) float v8f;

// Round-robin tournament pairing: D players, D-1 rounds, D/2 disjoint pairs/round.
__device__ __forceinline__ void rr_pair(int r, int k, int D, int& p, int& q) {
    int m = D - 1;
    if (k == 0) { p = m; q = r % m; }
    else        { p = (r + k) % m; q = (r - k + m) % m; }
    if (p > q) { int t = p; p = q; q = t; }
}

// Wave-parallel cyclic Jacobi eigensolver on DxD symmetric A (in LDS).
// Produces eigenvalues on diag(A) and eigenvectors in U (columns).
template<int D>
__device__ void jacobi_eigh(float* A, float* U, float* cb, float* sb,
                            int lane, int sweeps) {
    for (int t = lane; t < D * D; t += 32)
        U[t] = ((t / D) == (t % D)) ? 1.f : 0.f;
    __syncthreads();
    const int NP = D / 2;
    for (int sw = 0; sw < sweeps; ++sw) {
        for (int r = 0; r < D - 1; ++r) {
            // one lane per pair computes the rotation from pre-update A
            if (lane < NP) {
                int p, q; rr_pair(r, lane, D, p, q);
                float app = A[p * D + p], aqq = A[q * D + q], apq = A[p * D + q];
                float c = 1.f, s = 0.f;
                if (fabsf(apq) > 1e-12f) {
                    float tau = (aqq - app) / (2.f * apq);
                    float tt  = ((tau >= 0.f) ? 1.f : -1.f) /
                                (fabsf(tau) + sqrtf(1.f + tau * tau));
                    c = rsqrtf(1.f + tt * tt);
                    s = tt * c;
                }
                cb[lane] = c; sb[lane] = s;
            }
            __syncthreads();
            // phase 1: column updates  A <- A*G ,  U <- U*G  (pairs disjoint)
            for (int t = lane; t < NP * D; t += 32) {
                int k = t / D, i = t % D;
                int p, q; rr_pair(r, k, D, p, q);
                float c = cb[k], s = sb[k];
                float aip = A[i * D + p], aiq = A[i * D + q];
                A[i * D + p] = c * aip - s * aiq;
                A[i * D + q] = s * aip + c * aiq;
                float uip = U[i * D + p], uiq = U[i * D + q];
                U[i * D + p] = c * uip - s * uiq;
                U[i * D + q] = s * uip + c * uiq;
            }
            __syncthreads();
            // phase 2: row updates  A <- G^T*A
            for (int t = lane; t < NP * D; t += 32) {
                int k = t / D, j = t % D;
                int p, q; rr_pair(r, k, D, p, q);
                float c = cb[k], s = sb[k];
                float apj = A[p * D + j], aqj = A[q * D + j];
                A[p * D + j] = c * apj - s * aqj;
                A[q * D + j] = s * apj + c * aqj;
            }
            __syncthreads();
        }
    }
}

__global__ __launch_bounds__(32)
void spdnet_kernel(const float* __restrict__ x,  const float* __restrict__ W1,
                   const float* __restrict__ W2, const float* __restrict__ Wlin,
                   const float* __restrict__ blin, float* __restrict__ out, int B) {
    __shared__ float sx[TLEN_];
    __shared__ float sW1[WIN_ * D1_];
    __shared__ float sY[NPAD_ * D1_];     // projected windows (493x16, zero-padded)
    __shared__ float sA[D1_ * D1_];       // S1 / Jacobi workspace
    __shared__ float sU[D1_ * D1_];       // eigenvectors of S1
    __shared__ float sR[D1_ * D1_];       // S1r after ReEig
    __shared__ float sT[D1_ * D2_];       // S1r * W2
    __shared__ float sB2[D2_ * D2_];      // S2 / Jacobi workspace
    __shared__ float sU2[D2_ * D2_];      // eigenvectors of S2
    __shared__ float sL[D2_ * D2_];       // LogEig output
    __shared__ float scs[D1_];            // column sums of Yraw
    __shared__ float sd[D1_];             // clamped eigenvalues
    __shared__ float sd2[D2_];            // log eigenvalues
    __shared__ float cb[8], sb[8];        // Jacobi rotation c/s per pair

    const int b = blockIdx.x;
    if (b >= B) return;
    const int lane = threadIdx.x;
    const int half = lane >> 4;           // 0: lanes 0-15, 1: lanes 16-31
    const int m    = lane & 15;

    // ---- stage x row + W1 into LDS (one HBM pass per sample) ----
    const float* xr = x + (size_t)b * TLEN_;
    __builtin_prefetch(xr + lane * 16, 0, 1);     // global_prefetch_b8
    for (int i = lane; i < TLEN_; i += 32)       sx[i]  = xr[i];
    for (int i = lane; i < WIN_ * D1_; i += 32)  sW1[i] = W1[i];
    __syncthreads();

    // ---- Yraw[n,i] = sum_w x[n+w] * W1[w,i]  (fold W1 before covariance) ----
    for (int t = lane; t < NPAD_ * D1_; t += 32) {
        int n = t >> 4, i = t & 15;
        float a = 0.f;
        if (n < NWIN_) {
            #pragma unroll
            for (int w = 0; w < WIN_; ++w) a += sx[n + w] * sW1[w * D1_ + i];
        }
        sY[t] = a;                                // rows >= NWIN_ are zero
    }
    __syncthreads();

    // column sums for mean-centering correction
    if (lane < D1_) {
        float s = 0.f;
        for (int n = 0; n < NWIN_; ++n) s += sY[n * D1_ + lane];
        scs[lane] = s;
    }
    __syncthreads();

    // ---- R = Yraw^T * Yraw via V_WMMA_F32_16X16X4_F32 (K streamed in 4s) ----
    // For a symmetric product the 16x4 A-frag and 4x16 B-frag hold identical
    // register data under the CDNA5 lane layouts, so one LDS read feeds both.
    v8f acc = {0.f, 0.f, 0.f, 0.f, 0.f, 0.f, 0.f, 0.f};
    for (int n0 = 0; n0 < NPAD_; n0 += 4) {
        int n = n0 + 2 * half;                    // vgpr0: K = 2*half, vgpr1: K+1
        v2f f;
        f.x = sY[n * D1_ + m];
        f.y = sY[(n + 1) * D1_ + m];
        acc = __builtin_amdgcn_wmma_f32_16x16x4_f32(
                  false, f, false, f, (short)0, acc, false, false);
    }
    // S1 = (R - csum_i*csum_j/N) / (N-1); C/D layout: vgpr r -> row r+8*half, col m
    {
        const float invN   = 1.f / (float)NWIN_;
        const float invNm1 = 1.f / (float)(NWIN_ - 1);
        float cj = scs[m];
        #pragma unroll
        for (int r = 0; r < 8; ++r) {
            int Mi = r + 8 * half;
            sA[Mi * D1_ + m] = (acc[r] - scs[Mi] * cj * invN) * invNm1;
        }
    }
    __syncthreads();

    // ---- eigh(S1), 16x16 ----
    jacobi_eigh<D1_>(sA, sU, cb, sb, lane, 10);
    if (lane < D1_) sd[lane] = fmaxf(sA[lane * D1_ + lane], REEIG_EPS_);
    __syncthreads();

    // ---- ReEig reconstruction S1r = (U * diag(d)) * U^T via WMMA ----
    v8f acc2 = {0.f, 0.f, 0.f, 0.f, 0.f, 0.f, 0.f, 0.f};
    for (int k0 = 0; k0 < D1_; k0 += 4) {
        int k = k0 + 2 * half;
        v2f bf, af;
        bf.x = sU[m * D1_ + k];                   // B-frag: (U^T)[k][j=m]
        bf.y = sU[m * D1_ + k + 1];
        af.x = bf.x * sd[k];                      // A-frag: U[m][k]*d[k]
        af.y = bf.y * sd[k + 1];
        acc2 = __builtin_amdgcn_wmma_f32_16x16x4_f32(
                   false, af, false, bf, (short)0, acc2, false, false);
    }
    #pragma unroll
    for (int r = 0; r < 8; ++r) sR[(r + 8 * half) * D1_ + m] = acc2[r];
    __syncthreads();

    // ---- S2 = W2^T * S1r * W2  (16->8, tiny: VALU) ----
    for (int t = lane; t < D1_ * D2_; t += 32) {
        int i = t >> 3, c = t & 7;
        float a = 0.f;
        #pragma unroll
        for (int j = 0; j < D1_; ++j) a += sR[i * D1_ + j] * W2[j * D2_ + c];
        sT[t] = a;
    }
    __syncthreads();
    for (int t = lane; t < D2_ * D2_; t += 32) {
        int a2 = t >> 3, c = t & 7;
        float s = 0.f;
        #pragma unroll
        for (int i = 0; i < D1_; ++i) s += W2[i * D2_ + a2] * sT[i * D2_ + c];
        sB2[t] = s;
    }
    __syncthreads();

    // ---- eigh(S2), 8x8 ----
    jacobi_eigh<D2_>(sB2, sU2, cb, sb, lane, 8);
    if (lane < D2_) sd2[lane] = logf(fmaxf(sB2[lane * D2_ + lane], 1e-30f));
    __syncthreads();

    // ---- LogEig: L = U2 * diag(log ev) * U2^T ----
    for (int t = lane; t < D2_ * D2_; t += 32) {
        int i = t >> 3, j = t & 7;
        float a = 0.f;
        #pragma unroll
        for (int k = 0; k < D2_; ++k)
            a += sU2[i * D2_ + k] * sd2[k] * sU2[j * D2_ + k];
        sL[t] = a;
    }
    __syncthreads();

    // ---- linear head: out[b,c] = sum_k L[k]*Wlin[c,k] + blin[c] ----
    if (lane < NCLS_) {
        float a = blin[lane];
        for (int k = 0; k < D2_ * D2_; ++k)
            a += sL[k] * Wlin[lane * (D2_ * D2_) + k];
        out[(size_t)b * NCLS_ + lane] = a;
    }
}

extern "C" void kernel_launch(void* const* d_in, const int* in_sizes, int n_in,
                              void* d_out, int out_size, void* d_ws, size_t ws_size,
                              hipStream_t stream) {
    const float* x    = (const float*)d_in[0];
    const float* W1   = (const float*)d_in[1];
    const float* W2   = (const float*)d_in[2];
    const float* Wlin = (const float*)d_in[3];
    const float* blin = (const float*)d_in[4];
    float* out = (float*)d_out;
    const int B = in_sizes[0] / TLEN_;            // 8192
    spdnet_kernel<<<B, 32, 0, stream>>>(x, W1, W2, Wlin, blin, out, B);
}